// GCN_29712583754273
// MI455X (gfx1250) — compile-verified
//
#include <hip/hip_runtime.h>

typedef __attribute__((ext_vector_type(2))) float v2f;
typedef __attribute__((ext_vector_type(8))) float v8f;

#define DFEAT 128
#define COL_TILES (DFEAT / 16)

// ---------------------------------------------------------------------------
// Utility: fill a float buffer with a constant
// ---------------------------------------------------------------------------
__global__ __launch_bounds__(256) void fill_f32(float* __restrict__ p, float v, int n) {
    int i = blockIdx.x * blockDim.x + threadIdx.x;
    if (i < n) p[i] = v;
}

// ---------------------------------------------------------------------------
// Degree count: one thread per edge, atomicAdd 1.0 to deg[dst]
// (deg buffer pre-filled with 1.0 => self-loop contribution)
// ---------------------------------------------------------------------------
__global__ __launch_bounds__(256) void deg_count(const int* __restrict__ dst,
                                                 float* __restrict__ deg, int e) {
    int i = blockIdx.x * blockDim.x + threadIdx.x;
    if (i < e) atomicAdd(&deg[dst[i]], 1.0f);
}

// deg -> deg^{-1/2} in place
__global__ __launch_bounds__(256) void deg_to_dinv(float* __restrict__ d, int n) {
    int i = blockIdx.x * blockDim.x + threadIdx.x;
    if (i < n) {
        float v = d[i];
        d[i] = (v > 0.0f) ? rsqrtf(v) : 0.0f;
    }
}

// ---------------------------------------------------------------------------
// Dense GEMM Y[nrows x 128] = X[nrows x 128] * W[128 x 128] (+bias, +relu)
// One wave32 per 16x16 output tile; K marched in steps of 4 via
// v_wmma_f32_16x16x4_f32 (full f32 precision).
//
// A fragment (16x4 f32): lane<16 -> M=lane, K={k,k+1}; lane>=16 -> K={k+2,k+3}
// B fragment (4x16 f32): lane<16 -> N=lane, K={k,k+1}; lane>=16 -> K={k+2,k+3}
// C/D (16x16 f32, 8 VGPRs): vi -> M = (lane>>4)*8 + i, N = lane&15
// fuse: bit0 = add bias, bit1 = relu
// ---------------------------------------------------------------------------
__global__ __launch_bounds__(256) void gemm_wmma_f32(const float* __restrict__ X,
                                                     const float* __restrict__ W,
                                                     const float* __restrict__ bias,
                                                     float* __restrict__ Y,
                                                     int nrows, int fuse) {
    int wave = blockIdx.x * (blockDim.x >> 5) + (threadIdx.x >> 5);
    int lane = threadIdx.x & 31;
    int rowTiles = nrows >> 4;
    int tm = wave / COL_TILES;
    int tn = wave % COL_TILES;
    if (tm >= rowTiles) return;   // uniform per wave: EXEC stays all-ones for WMMA

    const int half = lane >> 4;   // 0: K pair {0,1}; 1: K pair {2,3}
    const int l15  = lane & 15;
    const int row  = (tm << 4) + l15;   // A-matrix row for this lane
    const int col  = (tn << 4) + l15;   // B/D column for this lane

    const float* xrow = X + (size_t)row * DFEAT + half * 2;          // contiguous K pair
    const float* wcol = W + (size_t)(half * 2) * DFEAT + col;        // strided K pair

    v8f acc = {0.f, 0.f, 0.f, 0.f, 0.f, 0.f, 0.f, 0.f};
#pragma unroll
    for (int k = 0; k < DFEAT; k += 4) {
        v2f a = *(const v2f*)(xrow + k);         // X[row, k+2*half .. +1], 8B aligned
        v2f b;
        b.x = wcol[(size_t)k * DFEAT];           // W[k+2*half,   col]
        b.y = wcol[(size_t)k * DFEAT + DFEAT];   // W[k+2*half+1, col]
        acc = __builtin_amdgcn_wmma_f32_16x16x4_f32(
            /*neg_a=*/false, a, /*neg_b=*/false, b,
            /*c_mod=*/(short)0, acc, /*reuse_a=*/false, /*reuse_b=*/false);
    }

    float bval = (fuse & 1) ? bias[col] : 0.0f;
    float* yout = Y + (size_t)((tm << 4) + (half << 3)) * DFEAT + (tn << 4) + l15;
#pragma unroll
    for (int i = 0; i < 8; ++i) {
        float v = acc[i] + bval;
        if (fuse & 2) v = fmaxf(v, 0.0f);
        yout[(size_t)i * DFEAT] = v;
    }
}

// ---------------------------------------------------------------------------
// Edge aggregation: one wave32 per edge, each lane handles 4 features.
// agg[dst] += H[src] * dinv[src]*dinv[dst]   (f32 atomics, L2-resident)
// ---------------------------------------------------------------------------
__global__ __launch_bounds__(256) void scatter_edges(const float* __restrict__ H,
                                                     const int* __restrict__ src,
                                                     const int* __restrict__ dst,
                                                     const float* __restrict__ dinv,
                                                     float* __restrict__ agg, int e) {
    int wave = blockIdx.x * (blockDim.x >> 5) + (threadIdx.x >> 5);
    int lane = threadIdx.x & 31;
    if (wave >= e) return;
    int s = src[wave];
    int d = dst[wave];
    float norm = dinv[s] * dinv[d];
    const float4 h4 = *(const float4*)(H + (size_t)s * DFEAT + (lane << 2));
    float* o = agg + (size_t)d * DFEAT + (lane << 2);
    atomicAdd(o + 0, h4.x * norm);
    atomicAdd(o + 1, h4.y * norm);
    atomicAdd(o + 2, h4.z * norm);
    atomicAdd(o + 3, h4.w * norm);
}

// ---------------------------------------------------------------------------
// Finalize per node: out = agg + Hself*dinv^2 + bias  (+relu)
// One wave per node, float4 per lane. In-place (out may alias agg) is safe:
// each element read+written by the same thread.
// ---------------------------------------------------------------------------
__global__ __launch_bounds__(256) void finalize_node(const float* __restrict__ agg,
                                                     const float* __restrict__ Hself,
                                                     const float* __restrict__ dinv,
                                                     const float* __restrict__ bias,
                                                     float* __restrict__ out,
                                                     int n, int relu) {
    int t = blockIdx.x * blockDim.x + threadIdx.x;
    int node = t >> 5;
    if (node >= n) return;
    int off = (t & 31) << 2;
    float di = dinv[node];
    float w = di * di;
    float4 a = *(const float4*)(agg + (size_t)node * DFEAT + off);
    float4 h = *(const float4*)(Hself + (size_t)node * DFEAT + off);
    float4 b = *(const float4*)(bias + off);
    float4 r;
    r.x = a.x + h.x * w + b.x;
    r.y = a.y + h.y * w + b.y;
    r.z = a.z + h.z * w + b.z;
    r.w = a.w + h.w * w + b.w;
    if (relu) {
        r.x = fmaxf(r.x, 0.0f);
        r.y = fmaxf(r.y, 0.0f);
        r.z = fmaxf(r.z, 0.0f);
        r.w = fmaxf(r.w, 0.0f);
    }
    *(float4*)(out + (size_t)node * DFEAT + off) = r;
}

// ---------------------------------------------------------------------------
extern "C" void kernel_launch(void* const* d_in, const int* in_sizes, int n_in,
                              void* d_out, int out_size, void* d_ws, size_t ws_size,
                              hipStream_t stream) {
    const float* x  = (const float*)d_in[0];
    const int*   ei = (const int*)d_in[1];           // [2, E] flat: row 0 = src, row 1 = dst
    const float* W0 = (const float*)d_in[2];
    const float* b0 = (const float*)d_in[3];
    const float* W1 = (const float*)d_in[4];
    const float* b1 = (const float*)d_in[5];
    const float* Wv = (const float*)d_in[6];
    const float* bv = (const float*)d_in[7];
    const float* Wt = (const float*)d_in[8];
    const float* bt = (const float*)d_in[9];

    const int N = in_sizes[0] / DFEAT;
    const int E = in_sizes[1] / 2;
    const int* src = ei;
    const int* dst = ei + E;

    float* out_h = (float*)d_out;
    float* out_v = out_h + (size_t)N * DFEAT;
    float* out_t = out_h + 2 * (size_t)N * DFEAT;

    // Workspace: dinv [N, padded] | bufA [N*D] | bufB [N*D]  (~51.4 MB)
    float* wsf  = (float*)d_ws;
    size_t npad = ((size_t)N + 127) & ~(size_t)127;
    float* dinv = wsf;
    float* bufA = wsf + npad;
    float* bufB = bufA + (size_t)N * DFEAT;

    const int nd = N * DFEAT;
    const int tiles = (N / 16) * COL_TILES;          // waves per GEMM
    const int gemmBlocks = (tiles + 7) / 8;          // 8 waves / 256-thread block
    const int edgeBlocks = (E + 7) / 8;              // 8 edges / block
    dim3 blk(256);

    // --- degree -> dinv --------------------------------------------------
    fill_f32<<<(N + 255) / 256, blk, 0, stream>>>(dinv, 1.0f, N);      // self-loops
    deg_count<<<(E + 255) / 256, blk, 0, stream>>>(dst, dinv, E);
    deg_to_dinv<<<(N + 255) / 256, blk, 0, stream>>>(dinv, N);

    // --- layer 0: h0 = relu(Agg(x @ W0) + b0) ----------------------------
    gemm_wmma_f32<<<gemmBlocks, blk, 0, stream>>>(x, W0, nullptr, bufA, N, 0);
    fill_f32<<<(nd + 255) / 256, blk, 0, stream>>>(bufB, 0.0f, nd);
    scatter_edges<<<edgeBlocks, blk, 0, stream>>>(bufA, src, dst, dinv, bufB, E);
    finalize_node<<<((N * 32) + 255) / 256, blk, 0, stream>>>(bufB, bufA, dinv, b0, bufA, N, 1);

    // --- layer 1: h = Agg(h0 @ W1) + b1  (into d_out section 0) ----------
    gemm_wmma_f32<<<gemmBlocks, blk, 0, stream>>>(bufA, W1, nullptr, bufB, N, 0);
    fill_f32<<<(nd + 255) / 256, blk, 0, stream>>>(out_h, 0.0f, nd);
    scatter_edges<<<edgeBlocks, blk, 0, stream>>>(bufB, src, dst, dinv, out_h, E);
    finalize_node<<<((N * 32) + 255) / 256, blk, 0, stream>>>(out_h, bufB, dinv, b1, out_h, N, 0);

    // --- heads: x_vision = relu(h@Wv+bv), x_text = relu(h@Wt+bt) ---------
    gemm_wmma_f32<<<gemmBlocks, blk, 0, stream>>>(out_h, Wv, bv, out_v, N, 3);
    gemm_wmma_f32<<<gemmBlocks, blk, 0, stream>>>(out_h, Wt, bt, out_t, N, 3);
}